// SiluAndMulQuant_57939108823596
// MI455X (gfx1250) — compile-verified
//
#include <hip/hip_runtime.h>
#include <stdint.h>

// ---------------------------------------------------------------------------
// SiluAndMulQuant for MI455X (gfx1250, wave32)
//   x: (8192, 22016) f32  ->  q: (8192, 11008) int8, scale: (8192) f32,
//                             act_sum: (8192) f32
// HBM-bandwidth-bound streaming kernel (~812 MB traffic, ~35us floor at
// 23.3 TB/s). Data path: CDNA5 async global->LDS copies (ASYNCcnt,
// double-buffered per wave), register-resident out-tile (inputs read from
// HBM exactly once), packed int8 nontemporal stores.
// VALU trimmed below the memory roofline via v_exp_f32 + v_rcp_f32 (1-ulp,
// far inside int8 quantization tolerance).
// ---------------------------------------------------------------------------

#define D_COLS   11008                 // half of 22016
#define ROW_F4   (D_COLS / 4)          // 2752 float4 per half-row
#define TPB      256                   // 8 wave32s
#define FULL_IT  (ROW_F4 / TPB)        // 10 full float4 iterations
#define REM_F4   (ROW_F4 % TPB)        // 192 -> waves 0..5 do one extra iter
#define MAX_IT   (FULL_IT + 1)         // 11
#define QMAX     127.0f

// Low 32 bits of a generic pointer into __shared__ memory are the
// wave-relative LDS byte address.
__device__ __forceinline__ uint32_t lds_addr_of(const void* p) {
  return (uint32_t)(uintptr_t)p;
}

// CDNA5 async copy: global memory -> LDS, tracked by ASYNCcnt.
// VGLOBAL encoding: vdst = LDS byte address VGPR, vaddr = 64-bit global addr.
// Input is streamed once (721 MB >> 192 MB L2) -> non-temporal hint.
__device__ __forceinline__ void async_copy_b128(uint32_t lds_byte_addr,
                                                const void* gaddr) {
  asm volatile("global_load_async_to_lds_b128 %0, %1, off th:TH_LOAD_NT"
               :: "v"(lds_byte_addr), "v"((uint64_t)(uintptr_t)gaddr)
               : "memory");
}

// Per-wave in-order completion => waiting to <=N retires the oldest copies.
#define WAIT_ASYNCCNT(n) asm volatile("s_wait_asynccnt %0" :: "i"(n) : "memory")

// silu(x) = x * sigmoid(x), using 1-ulp hardware transcendentals:
//   v_exp_f32 (base-2) + v_rcp_f32. ~4 VALU ops per element instead of the
//   ~12 a correctly-rounded expf+divide would cost.
__device__ __forceinline__ float silu_fast(float x) {
#if __has_builtin(__builtin_amdgcn_exp2f) && __has_builtin(__builtin_amdgcn_rcpf)
  const float e = __builtin_amdgcn_exp2f(x * -1.442695041f); // exp(-x)
  return x * __builtin_amdgcn_rcpf(1.0f + e);
#else
  return x * (1.0f / (1.0f + __expf(-x)));
#endif
}

__global__ __launch_bounds__(TPB, 1)
void silu_mul_quant_kernel(const float* __restrict__ x,
                           int8_t*      __restrict__ qout,
                           float*       __restrict__ scale_out,
                           float*       __restrict__ sum_out,
                           int ncols_total) {
  const int t   = threadIdx.x;
  const int wid = t >> 5;                       // wave id 0..7
  const size_t row = blockIdx.x;

  const float*  gate = x + row * (size_t)ncols_total;
  const float4* g4   = (const float4*)gate;              // 16B aligned
  const float4* u4   = (const float4*)(gate + D_COLS);   // 11008*4B -> aligned

  // Double-buffered staging: [phase][gate/up][thread] = 16 KB.
  __shared__ float4 lbuf[2][2][TPB];

  // Wave-uniform iteration count: REM_F4 = 192 -> waves 0..5 get 11 iters.
  const int nit = (wid < (REM_F4 >> 5)) ? MAX_IT : FULL_IT;

  // Prime the pipeline with chunk 0 (all waves active for i=0).
  async_copy_b128(lds_addr_of(&lbuf[0][0][t]), &g4[t]);
  async_copy_b128(lds_addr_of(&lbuf[0][1][t]), &u4[t]);

  float4 outv[MAX_IT];
  float lsum = 0.0f;
  float lmax = 0.0f;

  #pragma unroll
  for (int i = 0; i < MAX_IT; ++i) {
    if (i < nit) {
      if (i + 1 < nit) {                        // wave-uniform condition
        const int vn = (i + 1) * TPB + t;
        async_copy_b128(lds_addr_of(&lbuf[(i + 1) & 1][0][t]), &g4[vn]);
        async_copy_b128(lds_addr_of(&lbuf[(i + 1) & 1][1][t]), &u4[vn]);
        WAIT_ASYNCCNT(2);                       // chunk i done, i+1 in flight
      } else {
        WAIT_ASYNCCNT(0);                       // drain final chunk
      }
      const float4 g = lbuf[i & 1][0][t];
      const float4 u = lbuf[i & 1][1][t];
      float4 o;
      o.x = silu_fast(g.x) * u.x;
      o.y = silu_fast(g.y) * u.y;
      o.z = silu_fast(g.z) * u.z;
      o.w = silu_fast(g.w) * u.w;
      outv[i] = o;
      lsum += (o.x + o.y) + (o.z + o.w);
      lmax = fmaxf(lmax, fmaxf(fmaxf(fabsf(o.x), fabsf(o.y)),
                               fmaxf(fabsf(o.z), fabsf(o.w))));
    }
  }

  // ---- block reduction: sum + absmax (wave32 shuffle tree, then LDS) ----
  #pragma unroll
  for (int off = 16; off > 0; off >>= 1) {
    lsum += __shfl_xor(lsum, off, 32);
    lmax  = fmaxf(lmax, __shfl_xor(lmax, off, 32));
  }
  __shared__ float ssum[TPB / 32];
  __shared__ float smax[TPB / 32];
  if ((t & 31) == 0) { ssum[wid] = lsum; smax[wid] = lmax; }
  __syncthreads();

  float rsum = 0.0f, rmax = 0.0f;
  #pragma unroll
  for (int w = 0; w < TPB / 32; ++w) {
    rsum += ssum[w];
    rmax  = fmaxf(rmax, smax[w]);
  }

  const float scale = rmax * (1.0f / QMAX);
  const float inv   = (rmax > 0.0f) ? (QMAX / rmax) : 0.0f;

  if (t == 0) {
    scale_out[row] = scale;
    sum_out[row]   = rsum;
  }

  // ---- quantize from registers, 4 int8 packed per 32-bit NT store ----
  uint32_t* qrow = (uint32_t*)(qout + row * (size_t)D_COLS);
  #pragma unroll
  for (int i = 0; i < MAX_IT; ++i) {
    if (i < nit) {
      const int v = i * TPB + t;
      const float4 o = outv[i];
      // rintf == round-half-even, matching jnp.round; clip to [-128, 127]
      const int q0 = (int)fminf(fmaxf(rintf(o.x * inv), -128.0f), 127.0f);
      const int q1 = (int)fminf(fmaxf(rintf(o.y * inv), -128.0f), 127.0f);
      const int q2 = (int)fminf(fmaxf(rintf(o.z * inv), -128.0f), 127.0f);
      const int q3 = (int)fminf(fmaxf(rintf(o.w * inv), -128.0f), 127.0f);
      const uint32_t packed = ((uint32_t)(q0 & 0xFF))       |
                              ((uint32_t)(q1 & 0xFF) << 8)  |
                              ((uint32_t)(q2 & 0xFF) << 16) |
                              ((uint32_t)(q3 & 0xFF) << 24);
      __builtin_nontemporal_store(packed, qrow + v);
    }
  }
}

extern "C" void kernel_launch(void* const* d_in, const int* in_sizes, int n_in,
                              void* d_out, int out_size, void* d_ws, size_t ws_size,
                              hipStream_t stream) {
  (void)n_in; (void)out_size; (void)d_ws; (void)ws_size;

  const float* x   = (const float*)d_in[0];
  const int ncols  = 2 * D_COLS;                  // 22016
  const int rows   = in_sizes[0] / ncols;         // 8192

  // Tuple outputs concatenated flat in return order:
  //   q (rows*D_COLS int8) || scale (rows f32) || act_sum (rows f32)
  int8_t* q      = (int8_t*)d_out;
  float*  scale  = (float*)((char*)d_out + (size_t)rows * D_COLS);
  float*  actsum = scale + rows;

  silu_mul_quant_kernel<<<rows, TPB, 0, stream>>>(x, q, scale, actsum, ncols);
}